// LatentDiffModel_74311524155485
// MI455X (gfx1250) — compile-verified
//
#include <hip/hip_runtime.h>
#include <hip/hip_bf16.h>
#include <math.h>

// Problem dims
#define BB 8
#define LL 1024
#define DD 64
#define KK 4096
#define EE 512
#define MM (BB * LL)   // 8192 rows

typedef __attribute__((ext_vector_type(2))) float v2f;
typedef __attribute__((ext_vector_type(8))) float v8f;

__device__ __forceinline__ v8f wmma4(v2f a, v2f b, v8f c) {
    // V_WMMA_F32_16X16X4_F32 : D(16x16 f32) = A(16x4) * B(4x16) + C
    return __builtin_amdgcn_wmma_f32_16x16x4_f32(
        /*neg_a=*/false, a, /*neg_b=*/false, b,
        /*c_mod=*/(short)0, c, /*reuse_a=*/false, /*reuse_b=*/false);
}

// -------------------------------------------------------------------------
// Kernel 1: cond[row][e] = sinPE(s_idx,e) + chain_emb[c_idx][e] + entity_emb[e_idx][e]
// One pair (sin,cos of same angle) per thread-iteration, float2 I/O.
// -------------------------------------------------------------------------
__global__ __launch_bounds__(256) void cond_kernel(
    const int* __restrict__ mask, const int* __restrict__ sres,
    const int* __restrict__ cidx, const int* __restrict__ eidx,
    const float* __restrict__ chain_emb, const float* __restrict__ entity_emb,
    float* __restrict__ cond) {
    const int row = blockIdx.x;             // 0..MM-1
    const int m = mask[row];
    const int s = m ? sres[row] : 2047;     // PAD_NUM
    const int c = m ? cidx[row] : 63;       // PAD_CHAIN
    const int e = m ? eidx[row] : 63;       // PAD_ENT
    const float2* ce2 = (const float2*)(chain_emb + (size_t)c * EE);
    const float2* ee2 = (const float2*)(entity_emb + (size_t)e * EE);
    float2* out2 = (float2*)(cond + (size_t)row * EE);
    const float pos = (float)s;
    const float kln = 9.210340371976184f / 512.0f;  // ln(10000)/512
    const int p = threadIdx.x;                       // 256 pairs per row
    float ang = pos * expf(-(float)(2 * p) * kln);
    float sn, cs;
    sincosf(ang, &sn, &cs);
    float2 cv = ce2[p], ev = ee2[p], o;
    o.x = sn + cv.x + ev.x;
    o.y = cs + cv.y + ev.y;
    out2[p] = o;
}

// -------------------------------------------------------------------------
// Kernel 2: c_sq[n] = sum_k codebook[n][k]^2
// -------------------------------------------------------------------------
__global__ __launch_bounds__(256) void csq_kernel(
    const float* __restrict__ codebook, float* __restrict__ c_sq) {
    int n = blockIdx.x * blockDim.x + threadIdx.x;
    if (n < KK) {
        const float4* r = (const float4*)(codebook + (size_t)n * DD);
        float s = 0.f;
#pragma unroll
        for (int k = 0; k < DD / 4; ++k) {
            float4 v = r[k];
            s += v.x * v.x + v.y * v.y + v.z * v.z + v.w * v.w;
        }
        c_sq[n] = s;
    }
}

// -------------------------------------------------------------------------
// Kernel 3: per 32-row tile
//   h(32x512) = tanh([latent|cond](32x576) @ [Wx_top;Wc](576x512) + b)   (WMMA f32)
//   pred(32x64) = (h @ W_out) * mask                                     (WMMA f32)
//   p_sq(32) = rowsum(pred^2)
// 256 threads = 8 waves. LDS: 64KB reused (A-chunk staging -> h tile -> pred tile)
// -------------------------------------------------------------------------
__global__ __launch_bounds__(256) void mlp_kernel(
    const float* __restrict__ latent, const float* __restrict__ cond,
    const int* __restrict__ mask,
    const float* __restrict__ W_x, const float* __restrict__ W_c,
    const float* __restrict__ bias, const float* __restrict__ W_out,
    float* __restrict__ pred, float* __restrict__ p_sq) {
    __shared__ __align__(16) float lds[16384];   // 64 KB
    const int tid  = threadIdx.x;
    const int lane = tid & 31;
    const int wave = tid >> 5;
    const int lr   = lane & 15;
    const int half = lane >> 4;
    const int r0   = blockIdx.x * 32;

    const int n0w = wave * 64;            // each wave owns 64 of 512 N-columns
    v8f acc[2][4];
    const v8f vzero = {0.f,0.f,0.f,0.f,0.f,0.f,0.f,0.f};
#pragma unroll
    for (int mt = 0; mt < 2; ++mt)
#pragma unroll
        for (int nt = 0; nt < 4; ++nt) acc[mt][nt] = vzero;

    float4* dst4 = (float4*)lds;

    // ---- chunk 0: A = latent rows (contiguous 2048 floats), W = W_x top ----
    {
        const float4* src4 = (const float4*)(latent + (size_t)r0 * DD);
        dst4[tid]       = src4[tid];
        dst4[tid + 256] = src4[tid + 256];
    }
    __syncthreads();
    for (int k4 = 0; k4 < 64; k4 += 4) {
        const float* wrow = W_x + (size_t)k4 * EE;
        v2f a0, a1;
        a0.x = lds[lr        * 64 + k4 + 2 * half];
        a0.y = lds[lr        * 64 + k4 + 2 * half + 1];
        a1.x = lds[(16 + lr) * 64 + k4 + 2 * half];
        a1.y = lds[(16 + lr) * 64 + k4 + 2 * half + 1];
        v2f bf[4];
#pragma unroll
        for (int nt = 0; nt < 4; ++nt) {
            int col = n0w + nt * 16 + lr;
            bf[nt].x = wrow[(size_t)(2 * half)     * EE + col];
            bf[nt].y = wrow[(size_t)(2 * half + 1) * EE + col];
        }
#pragma unroll
        for (int nt = 0; nt < 4; ++nt) {
            acc[0][nt] = wmma4(a0, bf[nt], acc[0][nt]);
            acc[1][nt] = wmma4(a1, bf[nt], acc[1][nt]);
        }
    }

    // ---- chunks 1..8: A = cond 64-wide slices, W = W_c ----
    for (int kc = 1; kc < 9; ++kc) {
        const int kcol4 = (kc - 1) * 16;         // float4 column offset in cond row
        __syncthreads();
        {
            // 512 float4: j -> row = j>>4, col4 = j&15 ; cond row stride = 128 float4
            const float4* csrc4 = (const float4*)cond;
            int j0 = tid, j1 = tid + 256;
            dst4[j0] = csrc4[(size_t)(r0 + (j0 >> 4)) * 128 + kcol4 + (j0 & 15)];
            dst4[j1] = csrc4[(size_t)(r0 + (j1 >> 4)) * 128 + kcol4 + (j1 & 15)];
        }
        __syncthreads();
        const float* wbase = W_c + (size_t)(kc - 1) * 64 * EE;
        for (int k4 = 0; k4 < 64; k4 += 4) {
            const float* wrow = wbase + (size_t)k4 * EE;
            v2f a0, a1;
            a0.x = lds[lr        * 64 + k4 + 2 * half];
            a0.y = lds[lr        * 64 + k4 + 2 * half + 1];
            a1.x = lds[(16 + lr) * 64 + k4 + 2 * half];
            a1.y = lds[(16 + lr) * 64 + k4 + 2 * half + 1];
            v2f bf[4];
#pragma unroll
            for (int nt = 0; nt < 4; ++nt) {
                int col = n0w + nt * 16 + lr;
                bf[nt].x = wrow[(size_t)(2 * half)     * EE + col];
                bf[nt].y = wrow[(size_t)(2 * half + 1) * EE + col];
            }
#pragma unroll
            for (int nt = 0; nt < 4; ++nt) {
                acc[0][nt] = wmma4(a0, bf[nt], acc[0][nt]);
                acc[1][nt] = wmma4(a1, bf[nt], acc[1][nt]);
            }
        }
    }
    __syncthreads();

    // tanh + bias epilogue -> h tile (32 x 512) in LDS
#pragma unroll
    for (int mt = 0; mt < 2; ++mt)
#pragma unroll
        for (int nt = 0; nt < 4; ++nt) {
            int col = n0w + nt * 16 + lr;
            float bb = bias[col];
#pragma unroll
            for (int i = 0; i < 8; ++i) {
                int row = mt * 16 + i + 8 * half;
                lds[row * 512 + col] = tanhf(acc[mt][nt][i] + bb);
            }
        }
    __syncthreads();

    // ---- phase 2: pred = h @ W_out (512x64); 8 waves -> 8 16x16 tiles ----
    const int mT = wave & 1;     // 2 row-tiles
    const int nT = wave >> 1;    // 4 col-tiles
    v8f acc2 = vzero;
    for (int k = 0; k < 512; k += 4) {
        v2f a, bfr;
        a.x = lds[(mT * 16 + lr) * 512 + k + 2 * half];
        a.y = lds[(mT * 16 + lr) * 512 + k + 2 * half + 1];
        int col = nT * 16 + lr;
        bfr.x = W_out[(size_t)(k + 2 * half)     * DD + col];
        bfr.y = W_out[(size_t)(k + 2 * half + 1) * DD + col];
        acc2 = wmma4(a, bfr, acc2);
    }
    __syncthreads();

    // mask and write pred tile to LDS (32 x 64)
    {
        int col = nT * 16 + lr;
#pragma unroll
        for (int i = 0; i < 8; ++i) {
            int row = mT * 16 + i + 8 * half;
            float mv = (float)mask[r0 + row];
            lds[row * 64 + col] = acc2[i] * mv;
        }
    }
    __syncthreads();

    // store pred (b128, coalesced) and p_sq
    {
        float4* pdst4 = (float4*)(pred + (size_t)r0 * DD);
        const float4* lsrc4 = (const float4*)lds;
        pdst4[tid]       = lsrc4[tid];
        pdst4[tid + 256] = lsrc4[tid + 256];
    }
    if (tid < 32) {
        float s = 0.f;
#pragma unroll
        for (int k = 0; k < 64; ++k) { float v = lds[tid * 64 + k]; s += v * v; }
        p_sq[r0 + tid] = s;
    }
}

// -------------------------------------------------------------------------
// Kernel 4: l2[row][n] = p_sq[row] + c_sq[n] - 2 * (pred @ codebook^T)
// 64x128 output tile per block; LDS-staged operands; nontemporal stores.
// -------------------------------------------------------------------------
__global__ __launch_bounds__(256) void dist_kernel(
    const float* __restrict__ pred, const float* __restrict__ codebook,
    const float* __restrict__ p_sq, const float* __restrict__ c_sq,
    float* __restrict__ out) {
    __shared__ __align__(16) float sA[64 * 64];    // pred tile   [row][k]
    __shared__ __align__(16) float sB[128 * 64];   // codebook    [n][k]
    const int tid  = threadIdx.x;
    const int lane = tid & 31;
    const int wave = tid >> 5;
    const int lr   = lane & 15;
    const int half = lane >> 4;
    const int r0 = blockIdx.x * 64;
    const int n0 = blockIdx.y * 128;

    {
        const float4* pA4 = (const float4*)(pred + (size_t)r0 * 64);
        float4* dA4 = (float4*)sA;
#pragma unroll
        for (int j = 0; j < 4; ++j) dA4[tid + j * 256] = pA4[tid + j * 256];
        const float4* pB4 = (const float4*)(codebook + (size_t)n0 * 64);
        float4* dB4 = (float4*)sB;
#pragma unroll
        for (int j = 0; j < 8; ++j) dB4[tid + j * 256] = pB4[tid + j * 256];
    }
    __syncthreads();

    const int mT  = wave & 3;          // 4 row-tiles of 16
    const int nT0 = (wave >> 2) * 4;   // 2 groups of 4 col-tiles
    v8f acc[4];
    const v8f vzero = {0.f,0.f,0.f,0.f,0.f,0.f,0.f,0.f};
#pragma unroll
    for (int j = 0; j < 4; ++j) acc[j] = vzero;

    for (int k = 0; k < 64; k += 4) {
        v2f a;
        a.x = sA[(mT * 16 + lr) * 64 + k + 2 * half];
        a.y = sA[(mT * 16 + lr) * 64 + k + 2 * half + 1];
#pragma unroll
        for (int j = 0; j < 4; ++j) {
            // B[k][n] = codebook[n][k] -> contiguous v2f per lane from sB
            int n = (nT0 + j) * 16 + lr;
            v2f bfr;
            bfr.x = sB[n * 64 + k + 2 * half];
            bfr.y = sB[n * 64 + k + 2 * half + 1];
            acc[j] = wmma4(a, bfr, acc[j]);
        }
    }

    // epilogue: p_sq + c_sq - 2*cross, streaming nontemporal stores
    float ps[8];
#pragma unroll
    for (int i = 0; i < 8; ++i) ps[i] = p_sq[r0 + mT * 16 + i + 8 * half];
#pragma unroll
    for (int j = 0; j < 4; ++j) {
        const int colg = n0 + (nT0 + j) * 16 + lr;
        const float cs = c_sq[colg];
#pragma unroll
        for (int i = 0; i < 8; ++i) {
            const int rowg = r0 + mT * 16 + i + 8 * half;
            float val = ps[i] + cs - 2.0f * acc[j][i];
            __builtin_nontemporal_store(val, &out[(size_t)rowg * KK + colg]);
        }
    }
}

// -------------------------------------------------------------------------
extern "C" void kernel_launch(void* const* d_in, const int* in_sizes, int n_in,
                              void* d_out, int out_size, void* d_ws, size_t ws_size,
                              hipStream_t stream) {
    (void)in_sizes; (void)n_in; (void)out_size; (void)ws_size;
    const float* latent     = (const float*)d_in[0];
    // d_in[1] = t, d_in[2] = y : unused by the reference
    const int*   mask       = (const int*)d_in[3];
    const int*   sres       = (const int*)d_in[4];
    const int*   cidx       = (const int*)d_in[5];
    const int*   eidx       = (const int*)d_in[6];
    const float* chain_emb  = (const float*)d_in[7];
    const float* entity_emb = (const float*)d_in[8];
    const float* W_x        = (const float*)d_in[9];
    const float* W_c        = (const float*)d_in[10];
    const float* bias       = (const float*)d_in[11];
    const float* W_out      = (const float*)d_in[12];
    const float* codebook   = (const float*)d_in[13];
    float* out = (float*)d_out;

    // Workspace layout (~18.9 MB, L2-resident)
    char* ws = (char*)d_ws;
    float* cond = (float*)ws;                                        // 8192*512
    float* pred = (float*)(ws + (size_t)MM * EE * 4);                // 8192*64
    float* psq  = (float*)(ws + (size_t)MM * EE * 4 + (size_t)MM * DD * 4);  // 8192
    float* csq  = psq + MM;                                          // 4096

    cond_kernel<<<MM, 256, 0, stream>>>(mask, sres, cidx, eidx,
                                        chain_emb, entity_emb, cond);
    csq_kernel<<<KK / 256, 256, 0, stream>>>(codebook, csq);
    mlp_kernel<<<MM / 32, 256, 0, stream>>>(latent, cond, mask,
                                            W_x, W_c, bias, W_out, pred, psq);
    dim3 grid(MM / 64, KK / 128);
    dist_kernel<<<grid, 256, 0, stream>>>(pred, codebook, psq, csq, out);
}